// RotaryPositionalEmbedding_88278757802277
// MI455X (gfx1250) — compile-verified
//
#include <hip/hip_runtime.h>

// ---------------------------------------------------------------------------
// RoPE on MI455X (gfx1250).
//
// Reference einsum is a per-position matmul against a block-diagonal 2x2
// rotation matrix R[pos]. Useful work is pure streaming: 256MB x in, 256MB
// out, ~0.5 FLOP/byte -> hard bandwidth-bound (~22us at 23.3 TB/s). WMMA
// (f32 16x16x4) would add 64x wasted MACs + 256MB of R traffic, so the
// CDNA5 features used are:
//   * GLOBAL_LOAD_ASYNC_TO_LDS_B32: per-lane gather of the scattered
//     cos/sin diagonal entries of R straight into LDS (ASYNCcnt path,
//     drained with s_wait_asynccnt).
//   * Non-temporal hints (th:TH_LOAD_NT / th:TH_STORE_NT) on the 512MB
//     zero-reuse x/out stream, so the ~32MB of R diagonal cachelines
//     (reused 8x across bh-blocks) stay resident in the 192MB L2.
//   * Full unroll of the 16-row stream so each wave keeps 16 x 512B loads
//     in flight (MLP needed to cover DRAM latency at 23.3 TB/s).
// ---------------------------------------------------------------------------

#define D_KD     128
#define SEQ      4096
#define BH       128        // BATCH * N_HEADS = 4 * 32
#define S_TILE   8          // positions per block == waves per block
#define BH_TILE  16         // bh rows each wave streams
#define THREADS  256        // 8 wave32 waves

typedef float v4f __attribute__((ext_vector_type(4)));

__global__ __launch_bounds__(THREADS)
void rope_fused_kernel(const float* __restrict__ x,
                       const int*   __restrict__ pos,
                       const float* __restrict__ R,
                       float*       __restrict__ out)
{
    // Interleaved {cos_i, sin_i} table for the block's 8 positions: 4KB.
    __shared__ __align__(16) float cs[S_TILE * D_KD];

    const int tid = threadIdx.x;
    const int s0  = blockIdx.x * S_TILE;   // position chunk
    const int bh0 = blockIdx.y * BH_TILE;  // (batch*head) chunk

    // ------------------------------------------------------------------
    // Stage rotation coefficients with the CDNA5 async gather-to-LDS path.
    // value j in [0,1024):  s_local = j/128, d = j%128, pair i = d/2.
    //   d even -> cos_i = R[p, 2i,   2i]
    //   d odd  -> sin_i = R[p, 2i+1, 2i]
    // Each thread issues 4 async b32 gathers (ASYNCcnt += 4). These use the
    // default RT temporal policy so the R diagonal lines cache in L2.
    // ------------------------------------------------------------------
#pragma unroll
    for (int k = 0; k < 4; ++k) {
        const int j    = tid * 4 + k;
        const int sl   = j >> 7;          // local position
        const int d    = j & 127;
        const int i2   = d & ~1;          // 2*i
        const int isin = d & 1;
        const int p    = pos[s0 + sl];

        const float* gaddr =
            R + ((size_t)p << 14) + (size_t)(i2 + isin) * D_KD + (size_t)i2;
        const unsigned lds_off = (unsigned)(unsigned long long)(const void*)&cs[j];

        // GV mode: 64-bit per-lane global address in a VGPR pair, LDS byte
        // address in VDST. Writes LDS directly, no VGPR landing zone.
        asm volatile("global_load_async_to_lds_b32 %0, %1, off"
                     :: "v"(lds_off), "v"(gaddr)
                     : "memory");
    }

    // Drain this wave's async copies, then make LDS visible block-wide.
    asm volatile("s_wait_asynccnt 0x0" ::: "memory");
    __syncthreads();

    // ------------------------------------------------------------------
    // Streaming phase: wave w owns position s0+w. Lane L covers
    // d = 4L..4L+3, i.e. rotation pairs i0 = 2L and i1 = 2L+1.
    // cs float4 = {cos_i0, sin_i0, cos_i1, sin_i1} -- loaded once,
    // reused across BH_TILE rows.
    // ------------------------------------------------------------------
    const int wave = tid >> 5;
    const int lane = tid & 31;
    const int s    = s0 + wave;

    const v4f csv = *(const v4f*)&cs[wave * D_KD + lane * 4];

    const size_t row_base   = ((size_t)bh0 * SEQ + (size_t)s) * D_KD + (size_t)lane * 4;
    const size_t row_stride = (size_t)SEQ * D_KD;   // distance between bh rows

#pragma unroll
    for (int r = 0; r < BH_TILE; ++r) {
        const size_t off = row_base + (size_t)r * row_stride;
#if __has_builtin(__builtin_nontemporal_load)
        const v4f v = __builtin_nontemporal_load((const v4f*)(x + off));
#else
        const v4f v = *(const v4f*)(x + off);
#endif
        v4f o;
        o.x = csv.x * v.x - csv.y * v.y;   // cos*x0 - sin*x1
        o.y = csv.y * v.x + csv.x * v.y;   // sin*x0 + cos*x1
        o.z = csv.z * v.z - csv.w * v.w;
        o.w = csv.w * v.z + csv.z * v.w;
#if __has_builtin(__builtin_nontemporal_store)
        __builtin_nontemporal_store(o, (v4f*)(out + off));
#else
        *(v4f*)(out + off) = o;
#endif
    }
}

extern "C" void kernel_launch(void* const* d_in, const int* in_sizes, int n_in,
                              void* d_out, int out_size, void* d_ws, size_t ws_size,
                              hipStream_t stream)
{
    (void)in_sizes; (void)n_in; (void)out_size; (void)d_ws; (void)ws_size;

    const float* x   = (const float*)d_in[0];   // [4,32,4096,128] f32
    const int*   pos = (const int*)  d_in[1];   // [4096]
    const float* R   = (const float*)d_in[2];   // [4096,128,128] f32
    float*       out = (float*)d_out;           // [4,32,4096,128] f32

    dim3 grid(SEQ / S_TILE, BH / BH_TILE);      // (512, 8) = 4096 blocks
    rope_fused_kernel<<<grid, THREADS, 0, stream>>>(x, pos, R, out);
}